// LightGCNEncoder_48352741819113
// MI455X (gfx1250) — compile-verified
//
#include <hip/hip_runtime.h>
#include <stdint.h>

#define EMB   64   // embedding dim (fixed by reference)
#define EMB4  16   // float4 chunks per row
#define N_LAYERS 3
#define TILE  256  // edges staged per block-tile

// ---------------------------------------------------------------------------
// Kernel 1: build x0 = concat(user, item), acc = x0, y = 0.  All float4.
// ---------------------------------------------------------------------------
__global__ __launch_bounds__(256) void lgcn_init(
    const float* __restrict__ user_emb,
    const float* __restrict__ item_emb,
    float* __restrict__ x,      // ws buffer A: current layer features
    float* __restrict__ y,      // ws buffer B: scatter destination (zeroed)
    float* __restrict__ acc,    // d_out: running sum of layer embeddings
    long nUser4, long nTot4)
{
    const float4* u4 = (const float4*)user_emb;
    const float4* i4 = (const float4*)item_emb;
    float4* x4 = (float4*)x;
    float4* y4 = (float4*)y;
    float4* a4 = (float4*)acc;
    const float4 z4 = make_float4(0.f, 0.f, 0.f, 0.f);

    long stride = (long)gridDim.x * blockDim.x;
    for (long i = (long)blockIdx.x * blockDim.x + threadIdx.x; i < nTot4; i += stride) {
        float4 v = (i < nUser4) ? u4[i] : i4[i - nUser4];
        x4[i] = v;
        a4[i] = v;
        y4[i] = z4;
    }
}

// ---------------------------------------------------------------------------
// LDS byte-offset of a __shared__ pointer: flat LDS addresses carry the
// wave-relative offset in their low 32 bits (CDNA5 aperture mapping).
// ---------------------------------------------------------------------------
__device__ __forceinline__ uint32_t lds_off(const void* p) {
    return (uint32_t)(uintptr_t)p;
}

// ---------------------------------------------------------------------------
// Kernel 2: scatter pass with async metadata pipeline.
//
// Block = 256 threads = 16 groups of 16 lanes; each group owns one edge at a
// time (lane c moves float4 chunk c of the 64-float row).
//
// Edge metadata (src, dst, w) for a 256-edge tile is staged into LDS with
// GLOBAL_LOAD_ASYNC_TO_LDS_B128 (ASYNCcnt path), double-buffered so tile t+1
// streams in while tile t is processed.  Staging roles are wave-uniform:
// waves 0-1 stage src, 2-3 dst, 4-5 w, 6-7 idle (1 async b128 per thread).
//
// Per edge: 3 LDS broadcast reads, one b128 gather from x[src], 4 native
// non-returning f32 atomics into y[dst] (resolve in L2: dest fits in 192MB),
// plus a global_prefetch of the next edge's gather row using the LDS-resident
// source index (no global index load on the critical path).
// ---------------------------------------------------------------------------
__global__ __launch_bounds__(256) void lgcn_scatter(
    const float* __restrict__ x,
    float* __restrict__ y,
    const int* __restrict__ src,
    const int* __restrict__ dst,
    const float* __restrict__ w,
    long nEdges, long nTiles)
{
    __shared__ int   sSrc[2][TILE];
    __shared__ int   sDst[2][TILE];
    __shared__ float sW  [2][TILE];

    const int tid  = threadIdx.x;
    const int grp  = tid >> 4;        // edge group within block [0,16)
    const int c    = tid & 15;        // float4 chunk within row  [0,16)
    const int role = tid >> 6;        // wave-uniform: 0:src 1:dst 2:w 3:idle
    const int chnk = tid & 63;        // 16-byte chunk within tile [0,64)

    const float4* x4 = (const float4*)x;

    // ---- async stage of one tile's metadata into LDS buffer `buf` ----
    auto stage = [&](long tile, int buf) {
        if (role < 3) {
            long e0 = tile * TILE + (long)chnk * 4;   // first edge of chunk
            if (e0 + 4 > nEdges) e0 = (nEdges >= 4) ? (nEdges - 4) : 0; // clamp
            const int* gbase = (role == 0) ? src : (role == 1) ? dst : (const int*)w;
            void* lptr = (role == 0) ? (void*)&sSrc[buf][chnk * 4]
                       : (role == 1) ? (void*)&sDst[buf][chnk * 4]
                                     : (void*)&sW[buf][chnk * 4];
            uint32_t lds  = lds_off(lptr);
            uint32_t voff = (uint32_t)(e0 * 4);       // byte offset into array
            asm volatile("global_load_async_to_lds_b128 %0, %1, %2"
                         :: "v"(lds), "v"(voff), "s"(gbase) : "memory");
        }
    };

    const long tile0   = blockIdx.x;
    const long tstride = gridDim.x;
    int buf = 0;

    if (tile0 < nTiles) stage(tile0, 0);

    for (long t = tile0; t < nTiles; t += tstride) {
        const long tn = t + tstride;
        if (tn < nTiles) {
            stage(tn, buf ^ 1);
            // tile t's loads complete in order; allow tile tn's to stay in flight
            asm volatile("s_wait_asynccnt 0x1" ::: "memory");
        } else {
            asm volatile("s_wait_asynccnt 0x0" ::: "memory");
        }
        __syncthreads();

        const long eBase = t * TILE;
        for (int k = 0; k < 16; ++k) {
            const int  eLocal = grp * 16 + k;
            const long e      = eBase + eLocal;
            if (e >= nEdges) break;

            const int   s  = sSrc[buf][eLocal];
            const int   d  = sDst[buf][eLocal];
            const float wt = sW[buf][eLocal];

            // prefetch next edge's gather row (index already in LDS)
            if (k < 15 && e + 1 < nEdges) {
                const int sn = sSrc[buf][eLocal + 1];
                __builtin_prefetch(&x[(long)sn * EMB + c * 4], 0, 0);
            }

            float4 v = x4[(long)s * EMB4 + c];
            float* p = y + (long)d * EMB + c * 4;
            unsafeAtomicAdd(p + 0, v.x * wt);
            unsafeAtomicAdd(p + 1, v.y * wt);
            unsafeAtomicAdd(p + 2, v.z * wt);
            unsafeAtomicAdd(p + 3, v.w * wt);
        }
        __syncthreads();   // all groups done with buf before it is restaged
        buf ^= 1;
    }
}

// ---------------------------------------------------------------------------
// Kernel 3: fused accumulate + zero (+ optional final mean).
//   acc += y;  if last layer: acc *= 1/(N_LAYERS+1);  xOld = 0 (becomes the
//   scatter destination of the next layer).
// ---------------------------------------------------------------------------
__global__ __launch_bounds__(256) void lgcn_accum(
    float* __restrict__ acc,
    const float* __restrict__ y,
    float* __restrict__ xZero,
    long n4, int finalScale)
{
    float4* a4 = (float4*)acc;
    const float4* y4 = (const float4*)y;
    float4* z4 = (float4*)xZero;
    const float4 zero4 = make_float4(0.f, 0.f, 0.f, 0.f);
    const float sc = 1.0f / (float)(N_LAYERS + 1);

    long stride = (long)gridDim.x * blockDim.x;
    for (long i = (long)blockIdx.x * blockDim.x + threadIdx.x; i < n4; i += stride) {
        float4 a = a4[i];
        float4 b = y4[i];
        a.x += b.x; a.y += b.y; a.z += b.z; a.w += b.w;
        if (finalScale) { a.x *= sc; a.y *= sc; a.z *= sc; a.w *= sc; }
        a4[i] = a;
        z4[i] = zero4;
    }
}

// ---------------------------------------------------------------------------
// Host orchestration (graph-capture safe: only kernel launches on `stream`).
// ---------------------------------------------------------------------------
extern "C" void kernel_launch(void* const* d_in, const int* in_sizes, int n_in,
                              void* d_out, int out_size, void* d_ws, size_t ws_size,
                              hipStream_t stream) {
    const float* user_emb = (const float*)d_in[0];   // [N_USERS, 64] f32
    const float* item_emb = (const float*)d_in[1];   // [N_ITEMS, 64] f32
    const int*   eidx     = (const int*)d_in[2];     // [2, E] int32 (src row, dst row)
    const float* eweight  = (const float*)d_in[3];   // [E] f32

    const long nUser  = (long)in_sizes[0] / EMB;
    const long nItem  = (long)in_sizes[1] / EMB;
    const long nNodes = nUser + nItem;
    const long nEdges = (long)in_sizes[3];

    const int* src = eidx;            // edge_index[0]
    const int* dst = eidx + nEdges;   // edge_index[1]

    // Workspace: two ping-pong node-feature buffers (128 MB each).
    float* x = (float*)d_ws;
    float* y = x + (size_t)nNodes * EMB;
    float* acc = (float*)d_out;       // running sum lives directly in d_out

    const long n4     = nNodes * EMB4;     // float4 elements per node buffer
    const long nUser4 = nUser * EMB4;
    const long nTiles = (nEdges + TILE - 1) / TILE;

    dim3 blk(256);
    dim3 gridElem(4096);                                   // elementwise passes
    long gScat = nTiles < 1536 ? nTiles : 1536;            // ~3+ tiles per block
    dim3 gridScat((unsigned)(gScat > 0 ? gScat : 1));

    lgcn_init<<<gridElem, blk, 0, stream>>>(user_emb, item_emb, x, y, acc,
                                            nUser4, n4);

    for (int l = 0; l < N_LAYERS; ++l) {
        lgcn_scatter<<<gridScat, blk, 0, stream>>>(x, y, src, dst, eweight,
                                                   nEdges, nTiles);
        lgcn_accum<<<gridElem, blk, 0, stream>>>(acc, y, x, n4,
                                                 (l == N_LAYERS - 1) ? 1 : 0);
        float* t = x; x = y; y = t;   // new features become next gather source
    }
}